// RegionLoss_14714557956578
// MI455X (gfx1250) — compile-verified
//
#include <hip/hip_runtime.h>
#include <math.h>

// Problem dimensions (fixed by the reference)
#define NB 32
#define NA 5
#define NC 20
#define NH 76
#define NW 76
#define NT 50
#define HW (NH * NW)            // 5776
#define CELLS (NB * NA * HW)    // 924160  (== 3610 * 256 exactly)
#define NCH (5 + NC)            // 25 channels per anchor

#define ACC_N 9
// workspace layout (float offsets)
#define WS_ACC  0                      // 16 floats of accumulators (9 used)
#define WS_MASK 16
#define WS_CM   (WS_MASK + CELLS)
#define WS_TX   (WS_CM   + CELLS)
#define WS_TY   (WS_TX   + CELLS)
#define WS_TW   (WS_TY   + CELLS)
#define WS_TH   (WS_TW   + CELLS)
#define WS_TID  (WS_TH   + CELLS)

typedef __attribute__((ext_vector_type(2))) float v2f;
typedef __attribute__((ext_vector_type(8))) float v8f;

__device__ __constant__ float c_aw[NA] = {1.3221f, 3.19275f, 5.05587f, 9.47112f, 11.2364f};
__device__ __constant__ float c_ah[NA] = {1.73145f, 4.00944f, 8.09892f, 4.84053f, 10.0071f};

__device__ __forceinline__ float sigmoidf(float v) {
    return 1.0f / (1.0f + __expf(-v));
}

// ---------------------------------------------------------------------------
// Kernel 1: init mask=0, conf_mask=1, zero the global accumulators.
// tx/ty/tw/th/tid need no init: they are only read where mask==1, and the
// assign kernel writes them at exactly those cells.
// ---------------------------------------------------------------------------
__global__ void k_init(float* __restrict__ ws) {
    int idx = blockIdx.x * blockDim.x + threadIdx.x;
    if (blockIdx.x == 0 && threadIdx.x < 16) ws[WS_ACC + threadIdx.x] = 0.0f;
    if (idx < CELLS) {
        ws[WS_MASK + idx] = 0.0f;
        ws[WS_CM + idx] = 1.0f;
    }
}

// ---------------------------------------------------------------------------
// Kernel 2: ignore pass — conf_mask.min(keep): write 0 where iou > thres.
// Only writes the constant 0 -> scatter races are benign.
// ---------------------------------------------------------------------------
__global__ void k_ignore(const float* __restrict__ tgt, float* __restrict__ ws) {
    int idx = blockIdx.x * blockDim.x + threadIdx.x;
    if (idx >= NB * NT) return;
    int b = idx / NT;
    const float* t = tgt + idx * 5;
    float gw = t[2] * NW, gh = t[3] * NH;
    int gi = (int)(t[0] * NW);
    int gj = (int)(t[1] * NH);
    float area_g = (gw + 1.0f) * (gh + 1.0f);
#pragma unroll
    for (int a = 0; a < NA; ++a) {
        float aw = c_aw[a], ah = c_ah[a];
        float inter = fmaxf(fminf(gw, aw) + 1.0f, 0.0f) * fmaxf(fminf(gh, ah) + 1.0f, 0.0f);
        float iou = inter / (area_g + (aw + 1.0f) * (ah + 1.0f) - inter + 1e-16f);
        if (iou > 0.5f) {
            ws[WS_CM + ((b * NA + a) * NH + gj) * NW + gi] = 0.0f;
        }
    }
}

// ---------------------------------------------------------------------------
// Kernel 3: assign pass — best anchor per target; set conf_mask=1, mask=1,
// tx/ty/tw/th, class id at the winning cell.
// ---------------------------------------------------------------------------
__global__ void k_assign(const float* __restrict__ tgt, float* __restrict__ ws) {
    int idx = blockIdx.x * blockDim.x + threadIdx.x;
    if (idx >= NB * NT) return;
    int b = idx / NT;
    const float* t = tgt + idx * 5;
    float gx = t[0] * NW, gy = t[1] * NH;
    float gw = t[2] * NW, gh = t[3] * NH;
    int gi = (int)gx, gj = (int)gy;
    float area_g = (gw + 1.0f) * (gh + 1.0f);
    int best = 0;
    float best_iou = -1.0f;
#pragma unroll
    for (int a = 0; a < NA; ++a) {
        float aw = c_aw[a], ah = c_ah[a];
        float inter = fmaxf(fminf(gw, aw) + 1.0f, 0.0f) * fmaxf(fminf(gh, ah) + 1.0f, 0.0f);
        float iou = inter / (area_g + (aw + 1.0f) * (ah + 1.0f) - inter + 1e-16f);
        if (iou > best_iou) { best_iou = iou; best = a; }  // strict > == argmax first
    }
    int cell = ((b * NA + best) * NH + gj) * NW + gi;
    ws[WS_CM  + cell] = 1.0f;
    ws[WS_MASK + cell] = 1.0f;
    ws[WS_TX  + cell] = gx - (float)gi;
    ws[WS_TY  + cell] = gy - (float)gj;
    ws[WS_TW  + cell] = __logf(gw / c_aw[best] + 1e-16f);
    ws[WS_TH  + cell] = __logf(gh / c_ah[best] + 1e-16f);
    ws[WS_TID + cell] = t[4];
}

// ---------------------------------------------------------------------------
// Kernel 4: streaming masked-MSE partials. One thread per cell (grid exact,
// no divergence at the WMMA -> EXEC all ones as the ISA requires).
// Class channels + target regressors are only touched where mask==1
// (<=1600 of 924160 cells) -> ~3x HBM traffic reduction.
// Wave reduction via v_wmma_f32_16x16x4_f32 with A=ones:
//   sum over all 8 D VGPRs of all lanes == 16 * sum of all B inputs,
//   independent of B's lane layout. Finalize divides by 16.
// ---------------------------------------------------------------------------
__global__ void k_loss(const float* __restrict__ out, float* __restrict__ ws) {
    int idx = blockIdx.x * blockDim.x + threadIdx.x;   // < CELLS by construction
    int i = idx % NW;
    int j = (idx / NW) % NH;
    int a = (idx / HW) % NA;
    int b = idx / (NA * HW);
    int base = ((b * NA + a) * NCH) * HW + j * NW + i;

    float px = out[base];
    float py = out[base + HW];
    float pw = out[base + 2 * HW];
    float ph = out[base + 3 * HW];
    float pc = out[base + 4 * HW];

    float m  = ws[WS_MASK + idx];
    float cm = ws[WS_CM + idx];

    // confidence terms (tconf == mask; mask is exactly 0/1)
    float ec  = sigmoidf(pc) - m;
    float acf = ec * ec * (cm - m);
    float act = ec * ec * m;

    float ax = 0.0f, ay = 0.0f, aw2 = 0.0f, ah2 = 0.0f, acls = 0.0f;
    if (m > 0.0f) {
        float ex = sigmoidf(px) - ws[WS_TX + idx];
        float ey = sigmoidf(py) - ws[WS_TY + idx];
        float ew = pw - ws[WS_TW + idx];
        float eh = ph - ws[WS_TH + idx];
        ax = ex * ex; ay = ey * ey; aw2 = ew * ew; ah2 = eh * eh;
        int id = (int)ws[WS_TID + idx];
#pragma unroll 4
        for (int c = 0; c < NC; ++c) {
            __builtin_prefetch(out + base + (6 + c) * HW, 0, 1);  // global_prefetch_b8
            float p = sigmoidf(out[base + (5 + c) * HW]);
            float e = p - ((c == id) ? 1.0f : 0.0f);
            acls += e * e;
        }
    }

    float part[ACC_N] = {m, cm, ax, ay, aw2, ah2, acf, act, acls};

    __shared__ float sacc[ACC_N];
    if (threadIdx.x < ACC_N) sacc[threadIdx.x] = 0.0f;
    __syncthreads();

    v2f ones = {1.0f, 1.0f};
    v8f czero = {};
#pragma unroll
    for (int q = 0; q < ACC_N; ++q) {
        v2f bv = {part[q], 0.0f};
        // D[m][n] = sum_k 1 * B[k][n]; sum of all D slots = 16 * wave-sum(part)
        v8f d = __builtin_amdgcn_wmma_f32_16x16x4_f32(
            false, ones, false, bv, (short)0, czero, false, false);
        float ls = d[0] + d[1] + d[2] + d[3] + d[4] + d[5] + d[6] + d[7];
        atomicAdd(&sacc[q], ls);   // ds_add_f32
    }
    __syncthreads();
    if (threadIdx.x < ACC_N) {
        atomicAdd(&ws[WS_ACC + threadIdx.x], sacc[threadIdx.x]);
    }
}

// ---------------------------------------------------------------------------
// Kernel 5: finalize (accumulators carry the 16x WMMA replication factor).
// ---------------------------------------------------------------------------
__global__ void k_final(const float* __restrict__ ws, float* __restrict__ out) {
    float A[ACC_N];
#pragma unroll
    for (int q = 0; q < ACC_N; ++q) A[q] = ws[WS_ACC + q] * (1.0f / 16.0f);
    float sm   = A[0];
    float scmf = A[1] - A[0];            // sum(conf_mask) - sum(mask)
    float dm   = fmaxf(sm, 1.0f);
    float loss = (A[2] + A[3] + A[4] + A[5]) / dm      // x,y,w,h
               + A[6] / fmaxf(scmf, 1.0f)              // noobj conf
               + A[7] / dm                             // obj conf
               + A[8] / fmaxf(sm * (float)NC, 1.0f);   // class
    out[0] = loss;
}

// ---------------------------------------------------------------------------
extern "C" void kernel_launch(void* const* d_in, const int* in_sizes, int n_in,
                              void* d_out, int out_size, void* d_ws, size_t ws_size,
                              hipStream_t stream) {
    const float* output = (const float*)d_in[0];   // (32,125,76,76) f32
    const float* target = (const float*)d_in[1];   // (32,50,5) f32
    float* ws = (float*)d_ws;
    float* op = (float*)d_out;

    k_init<<<(CELLS + 255) / 256, 256, 0, stream>>>(ws);
    k_ignore<<<(NB * NT + 255) / 256, 256, 0, stream>>>(target, ws);
    k_assign<<<(NB * NT + 255) / 256, 256, 0, stream>>>(target, ws);
    k_loss<<<CELLS / 256, 256, 0, stream>>>(output, ws);   // 3610 blocks, exact
    k_final<<<1, 1, 0, stream>>>(ws, op);
}